// BaseSegHead_58634893525507
// MI455X (gfx1250) — compile-verified
//
#include <hip/hip_runtime.h>
#include <hip/hip_bf16.h>
#include <limits.h>

// Problem constants (from the reference)
#define N_FEAT   300
#define FH       200
#define FW       200
#define FHW      (FH * FW)          // 40000
#define NCLS     80                 // 81 logits, last dropped
#define NSCORES  (N_FEAT * NCLS)    // 24000
#define NCAND    1000
#define NMS_THR_ 0.65f
#define NSEL     100
#define OH       1024
#define OW       1024
#define LBL_OFF  201.0f             // float(max(fH,fW)) + 1
#define SCALE_   (200.0f / 1024.0f) // 0.1953125 exact

typedef float v4f __attribute__((ext_vector_type(4)));

// d_out layout (floats): [0,100) labels | [100, 100+100*1024*1024) masks |
//                        scores (100) | batch_ids (100)
#define MASKS_BASE   ((size_t)NSEL)
#define MASKS_ELEMS  ((size_t)NSEL * OH * OW)
#define SCORES_BASE  (MASKS_BASE + MASKS_ELEMS)
#define BATCH_BASE   (SCORES_BASE + NSEL)

// ---------------------------------------------------------------- kernel 5
// Bilinear 200x200 -> 1024x1024 (half-pixel, edge-clamped == jax.image.resize
// upsampling), threshold > 0. Each block: one mask slot x 64 output rows;
// the <=14 needed source rows are staged into LDS with gfx1250 async
// load-to-LDS; output streamed with 128-bit non-temporal stores.
// 4 consecutive output columns span <1 source pixel (scale 5.12x), so their
// whole 2x2 tap footprint fits a 3-column window [b, b+2]: 6 LDS loads per
// row-pair instead of 16, taps picked with cndmask selects.
#define MAX_SRC_ROWS 16
__global__ void k_resize(const float* __restrict__ seg,
                         const int* __restrict__ sel_feats,
                         float* __restrict__ out) {
    __shared__ float smem[MAX_SRC_ROWS * FW];
    const int s = blockIdx.y;       // output slot 0..99
    const int g = blockIdx.x;       // row group 0..15 (64 rows each)
    const int t = threadIdx.x;      // 0..255
    const int feat = sel_feats[s];
    const int Y0 = g * 64;

    // Source row range needed by this row group.
    float cy_lo = ((float)Y0 + 0.5f) * SCALE_ - 0.5f;
    float cy_hi = ((float)Y0 + 63.5f) * SCALE_ - 0.5f;
    int rlo = (int)floorf(cy_lo); if (rlo < 0) rlo = 0; if (rlo > FH - 1) rlo = FH - 1;
    int rhi = (int)floorf(cy_hi) + 1; if (rhi < rlo) rhi = rlo; if (rhi > FH - 1) rhi = FH - 1;
    int cnt = rhi - rlo + 1; if (cnt > MAX_SRC_ROWS) cnt = MAX_SRC_ROWS;
    const float* src = seg + (size_t)feat * FHW + (size_t)rlo * FW;
    const int elems = cnt * FW;

#if defined(__HIP_DEVICE_COMPILE__) && __has_builtin(__builtin_amdgcn_global_load_async_to_lds_b32)
    for (int e = t; e < elems; e += 256) {
        __builtin_amdgcn_global_load_async_to_lds_b32(
            (__attribute__((address_space(1))) int*)(src + e),
            (__attribute__((address_space(3))) int*)(&smem[e]),
            0, 0);
    }
#if __has_builtin(__builtin_amdgcn_s_wait_asynccnt)
    __builtin_amdgcn_s_wait_asynccnt(0);
#else
    asm volatile("s_wait_asynccnt 0" ::: "memory");
#endif
#else
    for (int e = t; e < elems; e += 256) smem[e] = src[e];
#endif
    __syncthreads();

    // Each thread owns 4 CONSECUTIVE output columns [4t, 4t+4) -> one b128 store.
    int x0[4], x1[4];
    float wx[4];
#pragma unroll
    for (int k = 0; k < 4; ++k) {
        int X = 4 * t + k;
        float cx = ((float)X + 0.5f) * SCALE_ - 0.5f;
        float xf = floorf(cx);
        wx[k] = cx - xf;
        int a = (int)xf;
        int a0 = a;     if (a0 < 0) a0 = 0; if (a0 > FW - 1) a0 = FW - 1;
        int a1 = a + 1; if (a1 < 0) a1 = 0; if (a1 > FW - 1) a1 = FW - 1;
        x0[k] = a0; x1[k] = a1;
    }
    // 3-column source window + per-column tap selectors (precomputed once).
    const int b = x0[0];
    const int c1i = (b + 1 < FW) ? b + 1 : FW - 1;
    const int c2i = (b + 2 < FW) ? b + 2 : FW - 1;
    int sel0[4], sel1[4];
#pragma unroll
    for (int k = 0; k < 4; ++k) {
        sel0[k] = (x0[k] == b) ? 0 : 1;                                   // x0 in {b, b+1}
        sel1[k] = (x1[k] == b) ? 0 : ((x1[k] == b + 1) ? 1 : 2);          // x1 in {b, b+1, b+2}
    }

    const size_t obase = MASKS_BASE + (size_t)s * (OH * OW);
    for (int r = 0; r < 64; ++r) {
        int Y = Y0 + r;
        float cy = ((float)Y + 0.5f) * SCALE_ - 0.5f;
        float yf = floorf(cy);
        float wy = cy - yf;
        int a = (int)yf;
        int y0 = a;     if (y0 < 0) y0 = 0; if (y0 > FH - 1) y0 = FH - 1;
        int y1 = a + 1; if (y1 < 0) y1 = 0; if (y1 > FH - 1) y1 = FH - 1;
        int ly0 = y0 - rlo; if (ly0 < 0) ly0 = 0; if (ly0 > cnt - 1) ly0 = cnt - 1;
        int ly1 = y1 - rlo; if (ly1 < 0) ly1 = 0; if (ly1 > cnt - 1) ly1 = cnt - 1;
        const float* r0 = &smem[ly0 * FW];
        const float* r1 = &smem[ly1 * FW];
        float a0 = r0[b], a1 = r0[c1i], a2 = r0[c2i];   // 6 LDS loads total
        float b0 = r1[b], b1 = r1[c1i], b2 = r1[c2i];
        v4f o;
#pragma unroll
        for (int k = 0; k < 4; ++k) {
            float t0 = sel0[k] ? a1 : a0;
            float t1 = (sel1[k] == 0) ? a0 : ((sel1[k] == 1) ? a1 : a2);
            float u0 = sel0[k] ? b1 : b0;
            float u1 = (sel1[k] == 0) ? b0 : ((sel1[k] == 1) ? b1 : b2);
            float top = t0 + wx[k] * (t1 - t0);
            float bot = u0 + wx[k] * (u1 - u0);
            float v = top + wy * (bot - top);
            o[k] = (v > 0.0f) ? 1.0f : 0.0f;
        }
        v4f* dst = (v4f*)&out[obase + (size_t)Y * OW + (size_t)(4 * t)];
        __builtin_nontemporal_store(o, dst);
    }
}

// ---------------------------------------------------------------- kernel 1
// Per-mask: non_empty flag + bbox. float4 loads (row length 200 % 4 == 0,
// so each 16B load stays inside one row); parallel argmin/argmax via
// shared atomics, reproducing jnp.argmax-of-all-False == 0 semantics.
__global__ void k_mask_stats(const float* __restrict__ seg,
                             float* __restrict__ boxes,
                             int* __restrict__ non_empty) {
    __shared__ int rows[FH];
    __shared__ int cols[FW];
    __shared__ int s_any, sxmin, sxmax, symin, symax;
    const int n = blockIdx.x;
    const v4f* m4 = (const v4f*)(seg + (size_t)n * FHW);
    for (int i = threadIdx.x; i < FH; i += blockDim.x) { rows[i] = 0; cols[i] = 0; }
    if (threadIdx.x == 0) {
        s_any = 0; sxmin = INT_MAX; sxmax = -1; symin = INT_MAX; symax = -1;
    }
    __syncthreads();
    int any = 0;
    for (int q = threadIdx.x; q < FHW / 4; q += blockDim.x) {
        v4f v = m4[q];
        int r  = q / (FW / 4);           // source row
        int cb = (q % (FW / 4)) * 4;     // first column of this float4
        int f0 = v.x > 0.0f, f1 = v.y > 0.0f, f2 = v.z > 0.0f, f3 = v.w > 0.0f;
        if (f0 | f1 | f2 | f3) { rows[r] = 1; any = 1; }
        if (f0) cols[cb + 0] = 1;
        if (f1) cols[cb + 1] = 1;
        if (f2) cols[cb + 2] = 1;
        if (f3) cols[cb + 3] = 1;
    }
    if (any) s_any = 1;   // benign same-value race
    __syncthreads();
    for (int i = threadIdx.x; i < FW; i += blockDim.x)
        if (cols[i]) { atomicMin(&sxmin, i); atomicMax(&sxmax, i); }
    for (int i = threadIdx.x; i < FH; i += blockDim.x)
        if (rows[i]) { atomicMin(&symin, i); atomicMax(&symax, i); }
    __syncthreads();
    if (threadIdx.x == 0) {
        int x1 = (sxmax < 0) ? 0 : sxmin;
        int x2 = (sxmax < 0) ? FW : (sxmax + 1);
        int y1 = (symax < 0) ? 0 : symin;
        int y2 = (symax < 0) ? FH : (symax + 1);
        boxes[n * 4 + 0] = (float)x1;
        boxes[n * 4 + 1] = (float)y1;
        boxes[n * 4 + 2] = (float)x2;
        boxes[n * 4 + 3] = (float)y2;
        non_empty[n] = s_any;
    }
}

// ---------------------------------------------------------------- kernel 2
__global__ void k_scores(const float* __restrict__ cls,
                         const int* __restrict__ non_empty,
                         float* __restrict__ scores) {
    int i = blockIdx.x * blockDim.x + threadIdx.x;
    if (i >= NSCORES) return;
    int n = i / NCLS, c = i % NCLS;
    float x = cls[n * (NCLS + 1) + c];
    float s = 1.0f / (1.0f + expf(-x));
    scores[i] = non_empty[n] ? s : -1.0f;
}

// ---------------------------------------------------------------- kernel 3
// Stable top-1000 by rank counting: rank = #{j : s_j > s_i or (s_j==s_i, j<i)}
// == jax.lax.top_k ordering. float4 LDS tiles + exact early exit: once a
// thread's rank >= 1000 it stops accumulating (result stays >= 1000, and
// only ranks < 1000 are used).
__global__ void k_topk1000(const float* __restrict__ scores,
                           float* __restrict__ cand_scores,
                           int* __restrict__ cand_ids) {
    __shared__ v4f tile4[256];   // 1024 floats per tile
    const int i = blockIdx.x * blockDim.x + threadIdx.x;
    const float si = (i < NSCORES) ? scores[i] : 0.0f;
    const v4f* sc4 = (const v4f*)scores;
    int rank = 0;
    for (int base = 0; base < NSCORES; base += 1024) {
        int cnt = NSCORES - base; if (cnt > 1024) cnt = 1024;   // always mult of 4
        int cnt4 = cnt >> 2;
        __syncthreads();
        if (threadIdx.x < cnt4) tile4[threadIdx.x] = sc4[(base >> 2) + threadIdx.x];
        __syncthreads();
        if (i < NSCORES && rank < NCAND) {
            for (int j4 = 0; j4 < cnt4; ++j4) {
                v4f w = tile4[j4];
                int gj = base + 4 * j4;
#pragma unroll
                for (int k = 0; k < 4; ++k) {
                    float sj = w[k];
                    rank += ((sj > si) || (sj == si && (gj + k) < i)) ? 1 : 0;
                }
            }
        }
    }
    if (i < NSCORES && rank < NCAND) {
        cand_scores[rank] = si;
        cand_ids[rank]    = i;
    }
}

// ---------------------------------------------------------------- kernel 4
// Single block of 256 threads (8 waves -> cheap barriers), 4 candidates per
// thread: greedy NMS over 1000 sorted candidates + stable top-100.
__global__ void k_nms_topk100(const float* __restrict__ boxes,
                              const float* __restrict__ cand_scores,
                              const int* __restrict__ cand_ids,
                              float* __restrict__ out,
                              int* __restrict__ sel_feats) {
    __shared__ float bx1[NCAND], by1[NCAND], bx2[NCAND], by2[NCAND];
    __shared__ float cs[NCAND], ks[NCAND];
    __shared__ int ci[NCAND];
    __shared__ unsigned char keep[NCAND];
    const int t = threadIdx.x;
#pragma unroll
    for (int k = 0; k < 4; ++k) {
        int j = t + k * 256;
        if (j < NCAND) {
            int id = cand_ids[j];
            ci[j] = id;
            int feat = id / NCLS;
            int lbl  = id % NCLS;
            float off = (float)lbl * LBL_OFF;
            bx1[j] = boxes[feat * 4 + 0] + off;
            by1[j] = boxes[feat * 4 + 1] + off;
            bx2[j] = boxes[feat * 4 + 2] + off;
            by2[j] = boxes[feat * 4 + 3] + off;
            cs[j]  = cand_scores[j];
            keep[j] = 1;
        }
    }
    __syncthreads();
    // Register-cache this thread's candidate boxes + areas.
    float mx1[4], my1[4], mx2[4], my2[4], mar[4];
#pragma unroll
    for (int k = 0; k < 4; ++k) {
        int j = t + k * 256;
        if (j < NCAND) {
            mx1[k] = bx1[j]; my1[k] = by1[j]; mx2[k] = bx2[j]; my2[k] = by2[j];
            mar[k] = (mx2[k] - mx1[k]) * (my2[k] - my1[k]);
        }
    }
    for (int i = 0; i < NCAND; ++i) {
        if (keep[i]) {
            float ax1 = bx1[i], ay1 = by1[i], ax2 = bx2[i], ay2 = by2[i];
            float areaA = (ax2 - ax1) * (ay2 - ay1);
#pragma unroll
            for (int k = 0; k < 4; ++k) {
                int j = t + k * 256;
                if (j > i && j < NCAND && keep[j]) {
                    float ix1 = fmaxf(ax1, mx1[k]);
                    float iy1 = fmaxf(ay1, my1[k]);
                    float ix2 = fminf(ax2, mx2[k]);
                    float iy2 = fminf(ay2, my2[k]);
                    float inter = fmaxf(ix2 - ix1, 0.0f) * fmaxf(iy2 - iy1, 0.0f);
                    float uni = areaA + mar[k] - inter;
                    float iou = inter / fmaxf(uni, 1e-9f);
                    if (iou > NMS_THR_) keep[j] = 0;
                }
            }
        }
        __syncthreads();
    }
#pragma unroll
    for (int k = 0; k < 4; ++k) {
        int j = t + k * 256;
        if (j < NCAND) ks[j] = (keep[j] && cs[j] >= 0.0f) ? cs[j] : -__builtin_inff();
    }
    __syncthreads();
#pragma unroll
    for (int k = 0; k < 4; ++k) {
        int jt = t + k * 256;
        if (jt < NCAND) {
            float st = ks[jt];
            int rank = 0;
            for (int j = 0; j < NCAND; ++j) {
                float sj = ks[j];
                rank += ((sj > st) || (sj == st && j < jt)) ? 1 : 0;
            }
            if (rank < NSEL) {
                int id = ci[jt];
                out[rank]               = (float)(id % NCLS);   // sel_labels
                out[SCORES_BASE + rank] = st;                   // sel_scores
                out[BATCH_BASE + rank]  = 0.0f;                 // batch_ids
                sel_feats[rank]         = id / NCLS;
            }
        }
    }
}

// ---------------------------------------------------------------- launcher
extern "C" void kernel_launch(void* const* d_in, const int* in_sizes, int n_in,
                              void* d_out, int out_size, void* d_ws, size_t ws_size,
                              hipStream_t stream) {
    const float* cls = (const float*)d_in[0];   // (300, 81) f32
    const float* seg = (const float*)d_in[1];   // (300, 200, 200) f32
    float* out = (float*)d_out;

    char* ws = (char*)d_ws;
    float* boxes       = (float*)(ws + 0);        // 300*4 f32   = 4800 B
    int*   non_empty   = (int*)  (ws + 5120);     // 300 i32     = 1200 B
    float* scores      = (float*)(ws + 6400);     // 24000 f32   = 96000 B (16B aligned)
    float* cand_scores = (float*)(ws + 102656);   // 1000 f32    = 4000 B
    int*   cand_ids    = (int*)  (ws + 106752);   // 1000 i32    = 4000 B
    int*   sel_feats   = (int*)  (ws + 110848);   // 100 i32     = 400 B

    k_mask_stats<<<N_FEAT, 256, 0, stream>>>(seg, boxes, non_empty);
    k_scores<<<(NSCORES + 255) / 256, 256, 0, stream>>>(cls, non_empty, scores);
    k_topk1000<<<(NSCORES + 255) / 256, 256, 0, stream>>>(scores, cand_scores, cand_ids);
    k_nms_topk100<<<1, 256, 0, stream>>>(boxes, cand_scores, cand_ids, out, sel_feats);
    k_resize<<<dim3(16, NSEL), 256, 0, stream>>>(seg, sel_feats, out);
}